// TopkRouting_39960375722105
// MI455X (gfx1250) — compile-verified
//
#include <hip/hip_runtime.h>
#include <hip/hip_bf16.h>

typedef float v8f __attribute__((ext_vector_type(8)));
typedef float v2f __attribute__((ext_vector_type(2)));

#define BATCH 8
#define CCH   512
#define NHEAD 8
#define DHEAD 64
#define NTOK  1024
#define KTOP  4
#define C3    1536

// ---------------- kernel 0: zero the (poisoned) output ----------------
__global__ void zero_out_kernel(float* __restrict__ out, int n) {
    int i = blockIdx.x * blockDim.x + threadIdx.x;
    if (i < n) out[i] = 0.0f;
}

// ---------------- kernel 1: QKV projection ----------------
// qkv[b,n,j] = sum_c x[b,c,n] * W[c,j] + bias[j]
// each wave: one 16(M=tokens) x 64(N=cols) tile via V_WMMA_F32_16X16X4_F32.
// j-group g in [0,24): which = g/8 (0=q,1=k,2=v), head = g%8.
// q,v stored [b,h,n,d]; k stored PAIR-INTERLEAVED [b,h,d/2,n,2] so the
// attention kernel's B-fragments (K^T[dd,j], K^T[dd+1,j]) are one b64 load.
__global__ __launch_bounds__(128) void qkv_kernel(const float* __restrict__ x,
                                                  const float* __restrict__ Wqkv,
                                                  const float* __restrict__ bqkv,
                                                  float* __restrict__ qbuf,
                                                  float* __restrict__ kTbuf,
                                                  float* __restrict__ vbuf) {
    const int lane    = threadIdx.x & 31;
    const int waveId  = blockIdx.x * 4 + (threadIdx.x >> 5);   // 12288 waves total
    const int b       = waveId / (64 * 24);
    const int rem     = waveId % (64 * 24);
    const int mtile   = rem / 24;
    const int ng      = rem % 24;
    const int which   = ng >> 3;       // 0=q 1=k 2=v
    const int head    = ng & 7;
    const int m0      = mtile * 16;
    const int j0      = ng * 64;
    const int halfsel = lane >> 4;     // ISA A-layout: lanes>=16 hold K pair {k0+2,k0+3}
    const int lm      = lane & 15;

    v8f acc[4];
    #pragma unroll
    for (int t = 0; t < 4; ++t) {
        float bv = bqkv[j0 + 16 * t + lm];
        #pragma unroll
        for (int r = 0; r < 8; ++r) acc[t][r] = bv;
    }

    const float* xb = x + (size_t)b * CCH * NTOK;
    for (int k0 = 0; k0 < CCH; k0 += 4) {
        const int kk = k0 + 2 * halfsel;
        v2f a;                                    // A[m, k] = x[b, k, m]
        a.x = xb[(size_t)kk * NTOK + m0 + lm];
        a.y = xb[(size_t)(kk + 1) * NTOK + m0 + lm];
        #pragma unroll
        for (int t = 0; t < 4; ++t) {
            v2f bf;                               // B[k, j] = W[k, j]
            bf.x = Wqkv[(size_t)kk * C3 + j0 + 16 * t + lm];
            bf.y = Wqkv[(size_t)(kk + 1) * C3 + j0 + 16 * t + lm];
            acc[t] = __builtin_amdgcn_wmma_f32_16x16x4_f32(
                false, a, false, bf, (short)0, acc[t], false, false);
        }
    }

    const int bh = b * NHEAD + head;
    #pragma unroll
    for (int t = 0; t < 4; ++t) {
        #pragma unroll
        for (int r = 0; r < 8; ++r) {
            const int m  = m0 + r + 8 * halfsel;   // C/D layout: lanes 16-31 hold M+8
            const int jl = 16 * t + lm;            // 0..63 within head (this is dd for k/v)
            const float val = acc[t][r];
            if (which == 0) {
                qbuf[((size_t)bh * NTOK + m) * DHEAD + jl] = val;
            } else if (which == 1) {
                // interleaved K^T: [bh][jl/2][m][2] + (jl&1)
                kTbuf[((((size_t)bh * (DHEAD / 2) + (jl >> 1)) * NTOK + m) << 1) + (jl & 1)] = val;
            } else {
                vbuf[((size_t)bh * NTOK + m) * DHEAD + jl] = val;
            }
        }
    }
}

// ---------------- kernel 2: scores + top-4 + softmax + weighted V + reduce ----------------
// one block (256 thr = 8 waves) per (b, h, 16-row i-tile). 4096 blocks.
__global__ __launch_bounds__(256) void attn_kernel(const float* __restrict__ qbuf,
                                                   const float* __restrict__ kTbuf,
                                                   const float* __restrict__ vbuf,
                                                   float* __restrict__ out) {
    __shared__ float sc[16][260];          // 16x256 score chunk; stride 260 avoids bank conflicts
    __shared__ float candw[16][16][KTOP];  // per-(row, stripe) top-4 candidates
    __shared__ int   candi[16][16][KTOP];
    __shared__ float topw[16][KTOP];
    __shared__ int   topi[16][KTOP];
    __shared__ float out_acc[DHEAD];

    const int tid     = threadIdx.x;
    const int lane    = tid & 31;
    const int wv      = tid >> 5;          // wave id 0..7 (uniform per wave)
    const int halfsel = lane >> 4;
    const int lm      = lane & 15;

    const int bh    = blockIdx.x >> 6;
    const int itile = blockIdx.x & 63;
    const int b     = bh >> 3;
    const int h     = bh & 7;
    const int i0    = itile * 16;

    const size_t qb = (size_t)bh * NTOK * DHEAD;
    const size_t vb = qb;
    const float2* kT2 = (const float2*)(kTbuf + (size_t)bh * DHEAD * NTOK); // [d/2][n] float2

    if (tid < DHEAD) out_acc[tid] = 0.0f;

    // preload this i-tile's Q A-fragments once (reused for all 64 j-tiles)
    v2f afrag[16];
    {
        const float* qrow = qbuf + qb + (size_t)(i0 + lm) * DHEAD + 2 * halfsel;
        #pragma unroll
        for (int kk = 0; kk < 16; ++kk) {
            afrag[kk].x = qrow[kk * 4];
            afrag[kk].y = qrow[kk * 4 + 1];
        }
    }

    // every thread owns a 16-column stripe of one row and keeps a running top-4
    // ordered by (value desc, index asc) -- exactly jax.lax.top_k ordering.
    const int srow = tid >> 4;             // 0..15
    const int ssub = tid & 15;             // 0..15 -> columns ssub*16 .. ssub*16+15 each chunk
    float lw[KTOP];
    int   li[KTOP];
    #pragma unroll
    for (int p = 0; p < KTOP; ++p) { lw[p] = -__builtin_inff(); li[p] = 0x7fffffff; }

    for (int chunk = 0; chunk < 4; ++chunk) {
        const int jcbase = chunk * 256;
        // phase A: 8 waves WMMA the 16x256 score chunk into LDS (2 j-tiles each)
        #pragma unroll
        for (int tsub = 0; tsub < 2; ++tsub) {
            const int jl0 = wv * 32 + tsub * 16;
            const int jg  = jcbase + jl0;
            v8f acc;
            #pragma unroll
            for (int r = 0; r < 8; ++r) acc[r] = 0.0f;
            #pragma unroll
            for (int kk = 0; kk < 16; ++kk) {
                // B[dd,j], B[dd+1,j] = one float2 from interleaved K^T (dd = kk*4 + 2*halfsel)
                const float2 bp = kT2[(size_t)(kk * 2 + halfsel) * NTOK + jg + lm];
                v2f bf; bf.x = bp.x; bf.y = bp.y;
                acc = __builtin_amdgcn_wmma_f32_16x16x4_f32(
                    false, afrag[kk], false, bf, (short)0, acc, false, false);
            }
            #pragma unroll
            for (int r = 0; r < 8; ++r)
                sc[r + 8 * halfsel][jl0 + lm] = acc[r];
        }
        __syncthreads();
        // phase B: all 256 threads scan their 16-column stripe into private top-4
        #pragma unroll
        for (int c = 0; c < 16; ++c) {
            const int cl = ssub * 16 + c;
            const float v = sc[srow][cl];
            const int   j = jcbase + cl;
            #pragma unroll
            for (int p = 0; p < KTOP; ++p) {
                const bool better = (v > lw[p]) || (v == lw[p] && j < li[p]);
                if (better) {
                    #pragma unroll
                    for (int q = KTOP - 1; q > p; --q) { lw[q] = lw[q - 1]; li[q] = li[q - 1]; }
                    lw[p] = v; li[p] = j;
                    break;
                }
            }
        }
        __syncthreads();   // before next chunk overwrites sc
    }

    // publish per-stripe candidates, then one thread per row merges 64 of them
    #pragma unroll
    for (int p = 0; p < KTOP; ++p) { candw[srow][ssub][p] = lw[p]; candi[srow][ssub][p] = li[p]; }
    __syncthreads();

    if (tid < 16) {
        const int row = tid;
        float w4[KTOP]; int i4[KTOP];
        #pragma unroll
        for (int p = 0; p < KTOP; ++p) { w4[p] = -__builtin_inff(); i4[p] = 0x7fffffff; }
        for (int s = 0; s < 16; ++s) {
            #pragma unroll
            for (int cp = 0; cp < KTOP; ++cp) {
                const float v = candw[row][s][cp];
                const int   j = candi[row][s][cp];
                #pragma unroll
                for (int p = 0; p < KTOP; ++p) {
                    const bool better = (v > w4[p]) || (v == w4[p] && j < i4[p]);
                    if (better) {
                        #pragma unroll
                        for (int q = KTOP - 1; q > p; --q) { w4[q] = w4[q - 1]; i4[q] = i4[q - 1]; }
                        w4[p] = v; i4[p] = j;
                        break;
                    }
                }
            }
        }
        // softmax over the 4 kept logits (w4[0] is the max)
        const float mx = w4[0];
        float e[KTOP], s = 0.0f;
        #pragma unroll
        for (int p = 0; p < KTOP; ++p) { e[p] = __expf(w4[p] - mx); s += e[p]; }
        const float inv = 1.0f / s;
        #pragma unroll
        for (int p = 0; p < KTOP; ++p) { topw[row][p] = e[p] * inv; topi[row][p] = i4[p]; }
    }
    __syncthreads();

    // weighted V gather + multiply by q, reduce the 16 rows via LDS float atomics
    {
        const int row = tid >> 4;
        const int dd0 = (tid & 15) * 4;
        #pragma unroll
        for (int u = 0; u < 4; ++u) {
            const int dd = dd0 + u;
            float wvv = 0.0f;
            #pragma unroll
            for (int p = 0; p < KTOP; ++p)
                wvv += topw[row][p] * vbuf[vb + (size_t)topi[row][p] * DHEAD + dd];
            const float val = wvv * qbuf[qb + (size_t)(i0 + row) * DHEAD + dd];
            atomicAdd(&out_acc[dd], val);
        }
    }
    __syncthreads();
    // out.transpose(0,2,1).reshape(B,C): flat index = b*512 + dd*NHEAD + h
    if (tid < DHEAD)
        atomicAdd(&out[b * CCH + tid * NHEAD + h], out_acc[tid]);
}

// ---------------- launcher ----------------
extern "C" void kernel_launch(void* const* d_in, const int* in_sizes, int n_in,
                              void* d_out, int out_size, void* d_ws, size_t ws_size,
                              hipStream_t stream) {
    const float* x  = (const float*)d_in[0];   // (8, 512, 32, 32)
    const float* Wq = (const float*)d_in[1];   // (512, 1536)
    const float* bq = (const float*)d_in[2];   // (1536,)
    float* out = (float*)d_out;                // (8, 512)
    float* ws  = (float*)d_ws;                 // needs 3 * 16 MB = 48 MB

    const size_t per = (size_t)BATCH * NHEAD * NTOK * DHEAD;   // 4M floats each
    float* qbuf = ws;
    float* kT   = ws + per;                    // pair-interleaved K^T
    float* vbuf = ws + 2 * per;

    zero_out_kernel<<<(out_size + 255) / 256, 256, 0, stream>>>(out, out_size);
    // 12288 waves = 8 batches * 64 m-tiles * 24 j-groups; 4 waves/block
    qkv_kernel<<<3072, 128, 0, stream>>>(x, Wq, bq, qbuf, kT, vbuf);
    // 4096 blocks = 64 (b,h) * 64 i-tiles
    attn_kernel<<<4096, 256, 0, stream>>>(qbuf, kT, vbuf, out);
}